// _ParametricLayer_41420664603257
// MI455X (gfx1250) — compile-verified
//
#include <hip/hip_runtime.h>

#define B_   2
#define N_   2048
#define FIN_ 128
#define H_   8
#define F_   32

typedef __attribute__((ext_vector_type(2))) float v2f;
typedef __attribute__((ext_vector_type(4))) float v4f;
typedef __attribute__((ext_vector_type(8))) float v8f;

// ---------------------------------------------------------------------------
// Kernel 1: wa[fin][col] = sum_f w[fin, h*32+f] * a[sel*32+f]
//   col = sel*8 + h ; sel=0 -> a1 (s_j), sel=1 -> a2 (s_i). 128x16 f32 output.
// ---------------------------------------------------------------------------
__global__ __launch_bounds__(256) void prep_kernel(const float* __restrict__ w,
                                                   const float* __restrict__ a,
                                                   float* __restrict__ wa) {
    for (int t = threadIdx.x; t < FIN_ * 16; t += 256) {
        const int fin = t >> 4;
        const int col = t & 15;
        const int h   = col & 7;
        const int sel = col >> 3;
        const float* av = a + sel * F_;
        const float* wv = w + fin * (H_ * F_) + h * F_;
        float acc = 0.f;
#pragma unroll
        for (int f = 0; f < F_; ++f) acc = fmaf(wv[f], av[f], acc);
        wa[fin * 16 + col] = acc;
    }
}

// ---------------------------------------------------------------------------
// Kernel 2: S[4096][16] = x[4096][128] @ wa[128][16]  via V_WMMA_F32_16X16X4_F32
//   cols 0..7  = s_j per head, cols 8..15 = s_i per head (full f32 precision).
//   One wave32 per 16-row tile; K=128 in 32 wmma steps.
//   A 16x4 frag: lanes 0-15 rows, vgpr0/1 = K, lane-half adds K+2.
//   B 4x16 frag: N striped across lanes, same K split. D: VGPR r -> M=r / r+8.
// ---------------------------------------------------------------------------
__global__ __launch_bounds__(32) void score_kernel(const float* __restrict__ x,
                                                   const float* __restrict__ wa,
                                                   float* __restrict__ S) {
    const int row0  = blockIdx.x * 16;
    const int lane  = threadIdx.x;
    const int m     = lane & 15;           // A row / B-D column
    const int khalf = (lane >> 4) * 2;     // lanes 16-31 hold K+2, K+3

    v8f c = {0.f, 0.f, 0.f, 0.f, 0.f, 0.f, 0.f, 0.f};
    const float* arow = x + (size_t)(row0 + m) * FIN_;

#pragma unroll 4
    for (int k = 0; k < FIN_; k += 4) {
        v2f af, bf;
        af.x = arow[k + khalf];
        af.y = arow[k + khalf + 1];
        bf.x = wa[(k + khalf) * 16 + m];
        bf.y = wa[(k + khalf + 1) * 16 + m];
        // (neg_a, A, neg_b, B, c_mod, C, reuse_a, reuse_b)
        c = __builtin_amdgcn_wmma_f32_16x16x4_f32(false, af, false, bf,
                                                  (short)0, c, false, false);
    }

    const int mo = (lane >> 4) * 8;        // D: lanes 16-31 hold M = r + 8
#pragma unroll
    for (int r = 0; r < 8; ++r)
        S[(size_t)(row0 + r + mo) * 16 + m] = c[r];
}

// ---------------------------------------------------------------------------
// Kernel 3: rn[b,i,h] = 1 / sum_j g[b,i,j] * exp(lrelu(s_i[b,i,h]+s_j[b,j,h]))
//   g is ~95% zero (+diagonal): skip the 8 exps when g==0.
//   One block per (b,i); 256 threads stride j; LDS tree reduction.
// ---------------------------------------------------------------------------
__global__ __launch_bounds__(256) void norm_kernel(const float* __restrict__ g,
                                                   const float* __restrict__ S,
                                                   float* __restrict__ rn) {
    const int bi  = blockIdx.x;            // b*N + i
    const int tid = threadIdx.x;
    const int b   = bi >> 11;

    float si[H_];
#pragma unroll
    for (int h = 0; h < H_; ++h) si[h] = S[(size_t)bi * 16 + 8 + h];

    float acc[H_];
#pragma unroll
    for (int h = 0; h < H_; ++h) acc[h] = 0.f;

    const float* grow = g + (size_t)bi * N_;
    const float* Sb   = S + (size_t)(b << 11) * 16;

    for (int j = tid; j < N_; j += 256) {
        const float gv = grow[j];
        if (gv != 0.f) {
            const float* Sj = Sb + (size_t)j * 16;
#pragma unroll
            for (int h = 0; h < H_; ++h) {
                const float s = si[h] + Sj[h];
                const float t = fmaxf(s, 0.f) + 0.2f * fminf(s, 0.f);
                acc[h] = fmaf(gv, __expf(t), acc[h]);
            }
        }
    }

    __shared__ float red[256 * H_];
#pragma unroll
    for (int h = 0; h < H_; ++h) red[tid * H_ + h] = acc[h];
    __syncthreads();
    for (int sft = 128; sft > 0; sft >>= 1) {
        if (tid < sft) {
#pragma unroll
            for (int h = 0; h < H_; ++h)
                red[tid * H_ + h] += red[(tid + sft) * H_ + h];
        }
        __syncthreads();
    }
    if (tid < H_) rn[(size_t)bi * H_ + tid] = 1.0f / red[tid];
}

// ---------------------------------------------------------------------------
// Kernel 4: out[b,i,j,h] = exp(lrelu(s_i+s_j)) * rn[b,i,h]
//   256 MB streaming write -> non-temporal b128 stores (don't pollute L2,
//   which keeps S/rn resident). One block per (b,i, 256-wide j stripe).
// ---------------------------------------------------------------------------
__global__ __launch_bounds__(256) void out_kernel(const float* __restrict__ S,
                                                  const float* __restrict__ rn,
                                                  float* __restrict__ out) {
    const int blk = blockIdx.x;
    const int bi  = blk >> 3;                       // b*N + i
    const int j   = ((blk & 7) << 8) | threadIdx.x; // 0..2047
    const int b   = bi >> 11;

    float si[H_], rr[H_];
#pragma unroll
    for (int h = 0; h < H_; ++h) {
        si[h] = S[(size_t)bi * 16 + 8 + h];
        rr[h] = rn[(size_t)bi * H_ + h];
    }

    const float* Sj = S + (size_t)((b << 11) | j) * 16;
    float e[H_];
#pragma unroll
    for (int h = 0; h < H_; ++h) {
        const float s = si[h] + Sj[h];
        const float t = fmaxf(s, 0.f) + 0.2f * fminf(s, 0.f);
        e[h] = __expf(t) * rr[h];
    }

    v4f* dst = (v4f*)(out + ((size_t)bi * N_ + (size_t)j) * H_);
    v4f lo = {e[0], e[1], e[2], e[3]};
    v4f hi = {e[4], e[5], e[6], e[7]};
    __builtin_nontemporal_store(lo, dst);
    __builtin_nontemporal_store(hi, dst + 1);
}

// ---------------------------------------------------------------------------
extern "C" void kernel_launch(void* const* d_in, const int* in_sizes, int n_in,
                              void* d_out, int out_size, void* d_ws, size_t ws_size,
                              hipStream_t stream) {
    (void)in_sizes; (void)n_in; (void)out_size; (void)ws_size;
    const float* g = (const float*)d_in[0];   // (B, N, N)
    const float* x = (const float*)d_in[1];   // (B, N, F_IN)
    const float* w = (const float*)d_in[2];   // (F_IN, H*F)
    const float* a = (const float*)d_in[3];   // (2*F,)
    float* out = (float*)d_out;               // (B, N, N, H)

    float* wa = (float*)d_ws;                         // 128*16 f32   (8 KB)
    float* S  = wa + FIN_ * 16;                       // 4096*16 f32  (256 KB)
    float* rn = S + (size_t)B_ * N_ * 16;             // 4096*8 f32   (128 KB)

    prep_kernel <<<1, 256, 0, stream>>>(w, a, wa);
    score_kernel<<<(B_ * N_) / 16, 32, 0, stream>>>(x, wa, S);
    norm_kernel <<<B_ * N_, 256, 0, stream>>>(g, S, rn);
    out_kernel  <<<B_ * N_ * (N_ / 256), 256, 0, stream>>>(S, rn, out);
}